// VanillaRNN_89885075570786
// MI455X (gfx1250) — compile-verified
//
#include <hip/hip_runtime.h>
#include <hip/hip_bf16.h>
#include <math.h>

typedef __attribute__((ext_vector_type(2))) float v2f;
typedef __attribute__((ext_vector_type(4))) float v4f;
typedef __attribute__((ext_vector_type(8))) float v8f;

static constexpr int VOCAB  = 50257;
static constexpr int EMBED  = 512;
static constexpr int HIDDEN = 1024;
static constexpr int SEQT   = 2048;

// ---------------- tiled fp32 WMMA GEMM ----------------
// C[M,N] = A[M,K] @ B[N,K]^T (+ bias[N]); A rows optionally gathered via rowIdx.
// Block: 256 threads (8 waves). Block tile 64(M) x 128(N), K step 32.
// Wave w: M sub-tile 16*(w&3), N strip 64*(w>>2); 4 accumulators of 16x16.
#define BM 64
#define BN 128
#define BK 32
#define LDSP (BK + 1)   // padded LDS row stride (floats)

// --- CDNA5 async global->LDS path (ASYNCcnt), with sync fallback ---
#if defined(__has_builtin)
#if __has_builtin(__builtin_amdgcn_global_load_async_to_lds_b128) && \
    __has_builtin(__builtin_amdgcn_s_wait_asynccnt)
#define HAVE_ASYNC_LDS 1
#endif
#endif

#ifdef HAVE_ASYNC_LDS
#define AS1 __attribute__((address_space(1)))
#define AS3 __attribute__((address_space(3)))
typedef int v4i_vs __attribute__((vector_size(4 * sizeof(int))));  // matches builtin param
__device__ __forceinline__ void async_copy16(const float* g, float* l)
{
    // per-lane 16B copy from global address g to LDS address l; tracked by ASYNCcnt
    __builtin_amdgcn_global_load_async_to_lds_b128(
        (AS1 v4i_vs*)g, (AS3 v4i_vs*)l, 0, 0);
}
#endif

__global__ __launch_bounds__(256)
void gemm_abt_wmma(const float* __restrict__ A,
                   const int*   __restrict__ rowIdx,
                   const float* __restrict__ B,
                   const float* __restrict__ bias,
                   float* __restrict__ C,
                   int M, int N, int K)
{
#ifdef HAVE_ASYNC_LDS
    constexpr int NBUF = 2;       // double-buffered tiles (async pipeline)
#else
    constexpr int NBUF = 1;
#endif
    __shared__ float Alds[NBUF][BM][LDSP];
    __shared__ float Blds[NBUF][BN][LDSP];

    const int tid  = threadIdx.x;
    const int lane = tid & 31;
    const int wave = tid >> 5;
    const int wm   = wave & 3;    // M sub-tile index (0..3)
    const int wn   = wave >> 2;   // N half (0..1)

    const int m0 = blockIdx.y * BM;
    const int n0 = blockIdx.x * BN;

    v8f acc[4];
    #pragma unroll
    for (int j = 0; j < 4; ++j)
        #pragma unroll
        for (int r = 0; r < 8; ++r) acc[j][r] = 0.0f;

    // cooperative tile-load mapping: 256 threads, 8 floats (2x b128) per chunk
    const int lr = tid >> 2;        // 0..63
    const int lc = (tid & 3) * 8;   // 0,8,16,24

    // thread-invariant global source pointers (K advances via +k0)
    const size_t arow = rowIdx ? (size_t)rowIdx[m0 + lr] : (size_t)(m0 + lr);
    const float* aptr = A + arow * (size_t)K + (size_t)lc;
    // B rows clamped into range: OOB columns only feed accumulators that the
    // bounds-checked store masks out, so staging stays branch-free & uniform.
    int nb0 = n0 + lr;      if (nb0 > N - 1) nb0 = N - 1;
    int nb1 = n0 + lr + 64; if (nb1 > N - 1) nb1 = N - 1;
    const float* bptr0 = B + (size_t)nb0 * (size_t)K + (size_t)lc;
    const float* bptr1 = B + (size_t)nb1 * (size_t)K + (size_t)lc;

#ifdef HAVE_ASYNC_LDS
    auto issue_stage = [&](int buf, int k0) {          // 6 async b128 ops/thread
        async_copy16(aptr + k0,      &Alds[buf][lr][lc]);
        async_copy16(aptr + k0 + 4,  &Alds[buf][lr][lc + 4]);
        async_copy16(bptr0 + k0,     &Blds[buf][lr][lc]);
        async_copy16(bptr0 + k0 + 4, &Blds[buf][lr][lc + 4]);
        async_copy16(bptr1 + k0,     &Blds[buf][lr + 64][lc]);
        async_copy16(bptr1 + k0 + 4, &Blds[buf][lr + 64][lc + 4]);
    };
#else
    auto issue_stage = [&](int buf, int k0) {
        v4f a0 = *(const v4f*)(aptr + k0);
        v4f a1 = *(const v4f*)(aptr + k0 + 4);
        v4f b0 = *(const v4f*)(bptr0 + k0);
        v4f b1 = *(const v4f*)(bptr0 + k0 + 4);
        v4f c0 = *(const v4f*)(bptr1 + k0);
        v4f c1 = *(const v4f*)(bptr1 + k0 + 4);
        #pragma unroll
        for (int q = 0; q < 4; ++q) {
            Alds[buf][lr][lc + q]          = a0[q];
            Alds[buf][lr][lc + 4 + q]      = a1[q];
            Blds[buf][lr][lc + q]          = b0[q];
            Blds[buf][lr][lc + 4 + q]      = b1[q];
            Blds[buf][lr + 64][lc + q]     = c0[q];
            Blds[buf][lr + 64][lc + 4 + q] = c1[q];
        }
    };
#endif

    // f32 16x16x4 fragment layout:
    // A frag: lanes 0-15 -> M=lane, K=kk+{0,1}; lanes 16-31 -> M=lane-16, K=kk+{2,3}
    // B frag: mirrored (N in lanes, same K split).
    auto compute = [&](int buf) {
        const int koff = (lane < 16) ? 0 : 2;
        const int ml   = lane & 15;
        const float* Ap = &Alds[buf][16 * wm + ml][0];
        #pragma unroll
        for (int kk = 0; kk < BK; kk += 4) {
            v2f a;
            a.x = Ap[kk + koff];
            a.y = Ap[kk + koff + 1];
            #pragma unroll
            for (int j = 0; j < 4; ++j) {
                const float* Bp = &Blds[buf][64 * wn + 16 * j + ml][0];
                v2f b;
                b.x = Bp[kk + koff];
                b.y = Bp[kk + koff + 1];
                acc[j] = __builtin_amdgcn_wmma_f32_16x16x4_f32(
                    false, a, false, b, (short)0, acc[j], false, false);
            }
        }
    };

#ifdef HAVE_ASYNC_LDS
    // software pipeline: tile k+1 streams into LDS while tile k computes
    issue_stage(0, 0);
    int buf = 0;
    for (int k0 = 0; k0 < K; k0 += BK, buf ^= 1) {
        const bool has_next = (k0 + BK) < K;
        if (has_next) {
            issue_stage(buf ^ 1, k0 + BK);
            // async ops complete in order: <=6 outstanding means the 6 ops of
            // the CURRENT buffer have landed; the next 6 keep flying.
            __builtin_amdgcn_s_wait_asynccnt(6);
        } else {
            __builtin_amdgcn_s_wait_asynccnt(0);
        }
        __syncthreads();
        compute(buf);
        __syncthreads();   // all reads of `buf` done before it is re-staged
    }
#else
    for (int k0 = 0; k0 < K; k0 += BK) {
        issue_stage(0, k0);
        __syncthreads();
        compute(0);
        __syncthreads();
    }
#endif

    // ---- store: D VGPR r -> (M = r + 8*(lane>=16), N = lane&15) ----
    const int mlo = lane & 15;
    const int mhi = (lane >> 4) * 8;
    #pragma unroll
    for (int j = 0; j < 4; ++j) {
        const int col = n0 + 64 * wn + 16 * j + mlo;
        if (col < N) {
            const float bv = bias ? bias[col] : 0.0f;
            #pragma unroll
            for (int r = 0; r < 8; ++r) {
                const int m = m0 + 16 * wm + r + mhi;
                C[(size_t)m * (size_t)N + (size_t)col] = acc[j][r] + bv;
            }
        }
    }
}

// ---------------- W_hh transpose (so scan reads coalesced) ----------------
__global__ __launch_bounds__(256)
void transpose_1024(const float* __restrict__ in, float* __restrict__ out)
{
    const int idx = blockIdx.x * 256 + threadIdx.x;   // < 1024*1024
    const int i = idx >> 10;        // row of W_hh
    const int k = idx & 1023;       // col of W_hh (coalesced read)
    out[(size_t)k * HIDDEN + i] = in[idx];
}

// ---------------- sequential recurrence ----------------
// One 1024-thread workgroup (32 wave32s on one WGP). h lives in LDS.
// W_hhT (4 MB) stays L2-resident across all 2048 steps.
__global__ __launch_bounds__(1024)
void rnn_scan(const float* __restrict__ xp,
              const float* __restrict__ WhhT,
              const float* __restrict__ b_hh,
              float* __restrict__ hs,
              float* __restrict__ h_final)
{
    __shared__ float h[HIDDEN];
    const int i = threadIdx.x;
    h[i] = 0.0f;
    const float bi = b_hh[i];
    __syncthreads();

    float hn = 0.0f;
    for (int t = 0; t < SEQT; ++t) {
        float acc = xp[(size_t)t * HIDDEN + i] + bi;
        #pragma unroll 8
        for (int k = 0; k < HIDDEN; ++k)
            acc = fmaf(WhhT[(size_t)k * HIDDEN + i], h[k], acc);  // h[k]: LDS broadcast
        hn = tanhf(acc);
        __syncthreads();            // everyone finished reading old h
        h[i] = hn;
        hs[(size_t)t * HIDDEN + i] = hn;
        __syncthreads();
    }
    h_final[i] = hn;
}

extern "C" void kernel_launch(void* const* d_in, const int* in_sizes, int n_in,
                              void* d_out, int out_size, void* d_ws, size_t ws_size,
                              hipStream_t stream)
{
    (void)in_sizes; (void)n_in; (void)out_size; (void)ws_size;

    const int*   x     = (const int*)  d_in[0];
    const float* emb   = (const float*)d_in[1];
    const float* W_xh  = (const float*)d_in[2];
    const float* W_hh  = (const float*)d_in[3];
    const float* b_hh  = (const float*)d_in[4];
    const float* W_out = (const float*)d_in[5];
    const float* b_out = (const float*)d_in[6];

    float* out = (float*)d_out;
    float* ws  = (float*)d_ws;

    // workspace layout (floats): xp[T*H] | W_hhT[H*H] | hs[T*H]  (20 MB total)
    float* xp   = ws;
    float* WhhT = xp + (size_t)SEQT * HIDDEN;
    float* hs   = WhhT + (size_t)HIDDEN * HIDDEN;

    float* logits  = out;                              // [T, VOCAB]
    float* h_final = out + (size_t)SEQT * VOCAB;       // [1, HIDDEN]

    // 1) W_hhT = W_hh^T
    transpose_1024<<<(HIDDEN * HIDDEN) / 256, 256, 0, stream>>>(W_hh, WhhT);

    // 2) xp = emb[x] @ W_xh^T   (gathered A, no bias)
    dim3 g1((HIDDEN + BN - 1) / BN, SEQT / BM);
    gemm_abt_wmma<<<g1, 256, 0, stream>>>(emb, x, W_xh, nullptr, xp,
                                          SEQT, HIDDEN, EMBED);

    // 3) sequential scan -> hs, h_final
    rnn_scan<<<1, 1024, 0, stream>>>(xp, WhhT, b_hh, hs, h_final);

    // 4) logits = hs @ W_out^T + b_out
    dim3 g2((VOCAB + BN - 1) / BN, SEQT / BM);
    gemm_abt_wmma<<<g2, 256, 0, stream>>>(hs, nullptr, W_out, b_out, logits,
                                          SEQT, VOCAB, HIDDEN);
}